// FastPitch_10376640987416
// MI455X (gfx1250) — compile-verified
//
#include <hip/hip_runtime.h>
#include <hip/hip_bf16.h>

// FastPitch regulate_len on MI455X (gfx1250).
// The indicator-matrix einsum is exactly a row gather: each of the B*L output
// frames copies one 1536-byte enc_out row (or zeros past dec_len). Roofline:
// ~12.6 MB read (L2-resident) + ~50.3 MB write -> ~2.7 us at 23.3 TB/s.
// WMMA would do 512x redundant FLOPs on this 0/1 matmul, so the right CDNA5
// path is the async global<->LDS DMA engine (ASYNCcnt): each wave streams its
// rows cache->LDS->memory with one wait point, no VGPR data path.

#define BB 16
#define TT 512
#define DD 384
#define LL 2048
#define PACE_F 1.0f
#define ROW_BYTES (DD * 4)     // 1536 bytes per row
#define CHUNKS 3               // 1536 / (32 lanes * 16 B)
#define FRAMES_PER_BLOCK 16    // 8 waves x 2 frames each

// ---------------------------------------------------------------------------
// Kernel 1: per-batch cumsum of reps (wave32 scan) + dec_lens tail output.
// ws layout: int cum[BB][TT+1]
// ---------------------------------------------------------------------------
__global__ __launch_bounds__(32) void fp_scan_kernel(
    const int* __restrict__ durations,   // [BB*TT] (int32 per harness contract)
    int* __restrict__ cum,               // [BB*(TT+1)] workspace
    float* __restrict__ out_tail)        // [BB] dec_lens (float, tuple tail)
{
  const int b    = blockIdx.x;
  const int lane = threadIdx.x;          // wave32: 16 tokens per lane

  int vals[16];
  int lane_sum = 0;
  const int base = b * TT + lane * 16;
#pragma unroll
  for (int k = 0; k < 16; ++k) {
    const int d = durations[base + k];
    const int r = (int)floorf((float)d / PACE_F + 0.5f);  // == d for PACE=1
    vals[k] = r;
    lane_sum += r;
  }

  // inclusive wave scan of per-lane sums
  int incl = lane_sum;
#pragma unroll
  for (int off = 1; off < 32; off <<= 1) {
    const int n = __shfl_up(incl, off, 32);
    if (lane >= off) incl += n;
  }
  const int excl = incl - lane_sum;

  int* cb = cum + b * (TT + 1);
  if (lane == 0) cb[0] = 0;
  int run = excl;
#pragma unroll
  for (int k = 0; k < 16; ++k) {
    run += vals[k];
    cb[lane * 16 + k + 1] = run;
  }
  if (lane == 31) {
    const int total = incl;
    out_tail[b] = (float)(total < LL ? total : LL);
  }
}

// ---------------------------------------------------------------------------
// Kernel 2: gather. 8 waves/block, 2 frames/wave, one LDS tile per frame.
// Phase A: issue all 6 async global->LDS b128 loads per lane.
// Phase B: single s_wait_asynccnt 0.
// Phase C: issue all 6 async LDS->global b128 stores (or plain zero stores),
//          drained by S_ENDPGM's implicit wait-idle.
// ---------------------------------------------------------------------------
__global__ __launch_bounds__(256) void fp_gather_kernel(
    const float* __restrict__ enc,       // [BB*TT*DD]
    const int* __restrict__ cum,         // [BB*(TT+1)]
    float* __restrict__ out)             // [BB*LL*DD]
{
  __shared__ int cumLds[TT + 1];
  __shared__ __align__(16) unsigned char tiles[8 * 2 * ROW_BYTES];  // 24 KB

  const int tid  = threadIdx.x;
  const int wave = tid >> 5;
  const int lane = tid & 31;
  const int b         = blockIdx.x / (LL / FRAMES_PER_BLOCK);
  const int frameBase = (blockIdx.x % (LL / FRAMES_PER_BLOCK)) * FRAMES_PER_BLOCK;

  // Stage this batch's cumsum into LDS (shared by all 16 frames in block).
  const int* cb = cum + b * (TT + 1);
  for (int i = tid; i < TT + 1; i += 256) cumLds[i] = cb[i];
  __syncthreads();

  const int total = cumLds[TT];
  const float* encB = enc + (size_t)b * TT * DD;
  float*       outB = out + (size_t)b * LL * DD;

  // Per-wave frames (wave-uniform) and their covering tokens.
  int   l[2];
  int   live[2];
  const float* src[2];
#pragma unroll
  for (int f = 0; f < 2; ++f) {
    l[f]    = frameBase + f * 8 + wave;
    live[f] = (l[f] < total);
    int lo = 0, hi = TT;                 // cum[0]=0 <= l < cum[TT] when live
    if (live[f]) {
      while (hi - lo > 1) {
        const int mid = (lo + hi) >> 1;
        if (cumLds[mid] <= l[f]) lo = mid; else hi = mid;
      }
    }
    src[f] = encB + (size_t)lo * DD;     // unique covering token row
  }

  // Phase A: async DMA global -> LDS for both frames (up to 6 b128 per lane).
#pragma unroll
  for (int f = 0; f < 2; ++f) {
    if (!live[f]) continue;
    unsigned char* tile = tiles + (size_t)(wave * 2 + f) * ROW_BYTES;
    const unsigned char* s = (const unsigned char*)src[f];
#pragma unroll
    for (int k = 0; k < CHUNKS; ++k) {
      const int byteOff = (k * 32 + lane) * 16;
      const unsigned int       lds_addr = (unsigned int)(uintptr_t)(tile + byteOff);
      const unsigned long long gaddr    = (unsigned long long)(uintptr_t)(s + byteOff);
      asm volatile("global_load_async_to_lds_b128 %0, %1, off"
                   :: "v"(lds_addr), "v"(gaddr) : "memory");
    }
  }

  // Phase B: one wait — all loads must land in LDS before stores read it.
  asm volatile("s_wait_asynccnt 0x0" ::: "memory");

  // Phase C: async DMA LDS -> output rows; zero rows use plain b128 stores.
#pragma unroll
  for (int f = 0; f < 2; ++f) {
    float* dst = outB + (size_t)l[f] * DD;
    if (live[f]) {
      unsigned char* tile = tiles + (size_t)(wave * 2 + f) * ROW_BYTES;
#pragma unroll
      for (int k = 0; k < CHUNKS; ++k) {
        const int byteOff = (k * 32 + lane) * 16;
        const unsigned int       lds_addr = (unsigned int)(uintptr_t)(tile + byteOff);
        const unsigned long long gaddr =
            (unsigned long long)(uintptr_t)((unsigned char*)dst + byteOff);
        asm volatile("global_store_async_from_lds_b128 %0, %1, off"
                     :: "v"(gaddr), "v"(lds_addr) : "memory");
      }
    } else {
      // Frame beyond dec_len: reference mult row is all-zero -> write zeros
      // (d_out is poisoned, so zeros must be explicit).
      const float4 z = make_float4(0.f, 0.f, 0.f, 0.f);
      float4* d4 = (float4*)dst;
#pragma unroll
      for (int k = 0; k < CHUNKS; ++k) d4[k * 32 + lane] = z;
    }
  }
  // Outstanding async stores drain at S_ENDPGM (implicit wait-idle).
}

// ---------------------------------------------------------------------------
extern "C" void kernel_launch(void* const* d_in, const int* in_sizes, int n_in,
                              void* d_out, int out_size, void* d_ws, size_t ws_size,
                              hipStream_t stream) {
  (void)in_sizes; (void)n_in; (void)out_size; (void)ws_size;

  const int*   durations = (const int*)d_in[0];    // [BB*TT]
  const float* enc       = (const float*)d_in[1];  // [BB*TT*DD]
  // d_in[2] = mel_max_len scalar (2048); shapes fixed by setup_inputs.

  float* out = (float*)d_out;                      // enc_rep flat, then dec_lens
  int*   cum = (int*)d_ws;                         // BB*(TT+1) ints = 32.8 KB

  float* dec_lens_tail = out + (size_t)BB * LL * DD;

  fp_scan_kernel<<<BB, 32, 0, stream>>>(durations, cum, dec_lens_tail);
  fp_gather_kernel<<<BB * (LL / FRAMES_PER_BLOCK), 256, 0, stream>>>(enc, cum, out);
}